// GCNPredictor_81097572483640
// MI455X (gfx1250) — compile-verified
//
#include <hip/hip_runtime.h>
#include <hip/hip_bf16.h>

typedef __attribute__((ext_vector_type(2))) float v2f;
typedef __attribute__((ext_vector_type(8))) float v8f;

#define ND_   10000
#define NDIS_ 10000
#define E_    400000
#define DIM_  128
#define EPS_  1e-5f

// ---------------- workspace layout (float offsets) ----------------
// zeroed region first (one zero kernel covers it all)
#define OFF_OUTDEG   0L                     // 10000  -> pad 10240
#define OFF_INDEG    10240L                 // 10000  -> pad 10240
#define OFF_AGG      20480L                 // 10000*128 = 1,280,000
#define OFF_SUM1     1300480L               // 256
#define OFF_SSQ1     1300736L               // 256
#define OFF_SUM2     1300992L               // 128
#define OFF_SSQ2     1301120L               // 128
#define ZERO_N       1301248L               // end of zeroed region
// non-zeroed region
#define OFF_NORMOUT  1301248L               // 10000 -> pad 10240
#define OFF_NORMIN   1311488L               // 10000 -> pad 10240
#define OFF_DISH     1321728L               // 10000*256 = 2,560,000
#define OFF_Y1       3881728L               // 10000*256
#define OFF_Y2       6441728L               // 10000*128 = 1,280,000
#define OFF_C1       7721728L               // 256
#define OFF_SDIS     7721984L               // 10000

// ---------------- utility kernels ----------------
__global__ void zero_ws_kernel(float* p, long n) {
    long i = (long)blockIdx.x * blockDim.x + threadIdx.x;
    if (i < n) p[i] = 0.0f;
}

__global__ void degrees_kernel(const int* __restrict__ src, const int* __restrict__ dst,
                               float* __restrict__ outd, float* __restrict__ ind, int E) {
    int e = blockIdx.x * blockDim.x + threadIdx.x;
    if (e < E) {
        atomicAdd(&outd[src[e]], 1.0f);
        atomicAdd(&ind[dst[e]], 1.0f);
    }
}

__global__ void make_norm_kernel(const float* __restrict__ deg, float* __restrict__ nrm, int n) {
    int i = blockIdx.x * blockDim.x + threadIdx.x;
    if (i < n) nrm[i] = rsqrtf(fmaxf(deg[i], 1.0f));
}

// agg[dst[e]][f] += h_drug[src[e]][f] * norm_out[src[e]]
__global__ void scatter_agg_kernel(const float* __restrict__ h_drug, const int* __restrict__ src,
                                   const int* __restrict__ dst, const float* __restrict__ norm_out,
                                   float* __restrict__ agg, int E) {
    long idx = (long)blockIdx.x * blockDim.x + threadIdx.x;
    if (idx >= (long)E * DIM_) return;
    int e = (int)(idx >> 7);
    int f = (int)(idx & (DIM_ - 1));
    int s = src[e];
    float v = h_drug[(size_t)s * DIM_ + f] * norm_out[s];
    atomicAdd(&agg[(size_t)dst[e] * DIM_ + f], v);
}

// ---------------- FP32 WMMA GEMM: C[M,N] = (rowscale.A)[M,K] @ B[K,N] + bias ----------------
// one 16x16 C tile per wave32, K stepped by 4 via V_WMMA_F32_16X16X4_F32.
// A-frag layout (ISA 7.12.2): lanes 0-15 row M=m0+l; VGPR0 = K{0 | 2}, VGPR1 = K{1 | 3} (by half-wave)
// B-frag: lanes 0-15 col N=n0+l; VGPR0 = K{0 | 2}, VGPR1 = K{1 | 3}
// D: VGPR j -> row m0 + j + 8*half, col n0 + (lane&15)
__global__ void gemm16_wmma_kernel(const float* __restrict__ A, const float* __restrict__ B,
                                   const float* __restrict__ bias, const float* __restrict__ rowscale,
                                   float* __restrict__ C, int M, int N, int K) {
    const int lane = threadIdx.x & 31;
    const int wave = threadIdx.x >> 5;
    const int ntiles = N >> 4;
    const int mtiles = M >> 4;
    int tile = blockIdx.x * (blockDim.x >> 5) + wave;   // wave-uniform
    if (tile >= mtiles * ntiles) return;                // whole-wave exit: EXEC stays all-1s

    const int tm = tile / ntiles, tn = tile % ntiles;
    const int m0 = tm << 4, n0 = tn << 4;
    const int half = lane >> 4;        // 0: K{0,1}, 1: K{2,3}
    const int l = lane & 15;

    const float* Arow = A + (size_t)(m0 + l) * K;
    const float  scale = rowscale ? rowscale[m0 + l] : 1.0f;
    const float* Bcol = B + (n0 + l);

    v8f acc = {};
    for (int k = 0; k < K; k += 4) {
        v2f a, b;
        int ka = k + 2 * half;
        a.x = Arow[ka] * scale;
        a.y = Arow[ka + 1] * scale;
        b.x = Bcol[(size_t)ka * N];
        b.y = Bcol[(size_t)(ka + 1) * N];
        acc = __builtin_amdgcn_wmma_f32_16x16x4_f32(
            /*neg_a=*/false, a, /*neg_b=*/false, b,
            /*c_mod=*/(short)0, acc, /*reuse_a=*/false, /*reuse_b=*/false);
    }
    const float bn = bias ? bias[n0 + l] : 0.0f;
    #pragma unroll
    for (int j = 0; j < 8; ++j) {
        int row = m0 + j + half * 8;
        C[(size_t)row * N + n0 + l] = acc[j] + bn;
    }
}

// c1[n] = b1[n] + sum_k b_gcn[k] * W1[k*256 + n]   (constant drug-half contribution)
__global__ void compute_c1_kernel(const float* __restrict__ b_gcn, const float* __restrict__ W1,
                                  const float* __restrict__ b1, float* __restrict__ c1) {
    int n = threadIdx.x;               // 256 threads, 1 block
    float acc = b1[n];
    for (int k = 0; k < 256; ++k) acc += b_gcn[k] * W1[k * 256 + n];
    c1[n] = acc;
}

// weighted column sums for BN over the E-edge batch: w[d] = in_deg[d]
__global__ void bn_stats_kernel(const float* __restrict__ Y, const float* __restrict__ cnt,
                                float* __restrict__ sum, float* __restrict__ sumsq,
                                int M, int N, int rows_per_block) {
    int c = threadIdx.x;
    if (c >= N) return;
    int r0 = blockIdx.x * rows_per_block;
    int r1 = r0 + rows_per_block; if (r1 > M) r1 = M;
    float s = 0.0f, s2 = 0.0f;
    for (int r = r0; r < r1; ++r) {
        float w = cnt[r];
        float v = Y[(size_t)r * N + c];
        s += w * v;
        s2 += w * v * v;
    }
    atomicAdd(&sum[c], s);
    atomicAdd(&sumsq[c], s2);
}

// in-place: Y = relu((Y - m) * rsqrt(v + eps) * gamma + beta)
__global__ void bn_apply_relu_kernel(float* __restrict__ Y, const float* __restrict__ sum,
                                     const float* __restrict__ sumsq,
                                     const float* __restrict__ gamma, const float* __restrict__ beta,
                                     float invE, long total, int N) {
    long i = (long)blockIdx.x * blockDim.x + threadIdx.x;
    if (i >= total) return;
    int c = (int)(i % N);
    float m = sum[c] * invE;
    float var = sumsq[c] * invE - m * m;
    float x = (Y[i] - m) * rsqrtf(var + EPS_) * gamma[c] + beta[c];
    Y[i] = x > 0.0f ? x : 0.0f;
}

// per-disease: BN2 + relu + dot(W3) + b3 + sigmoid
__global__ void final_score_kernel(const float* __restrict__ Y2,
                                   const float* __restrict__ sum2, const float* __restrict__ sumsq2,
                                   const float* __restrict__ gamma2, const float* __restrict__ beta2,
                                   const float* __restrict__ W3, const float* __restrict__ b3,
                                   float invE, float* __restrict__ sdis, int M) {
    __shared__ float w3s[DIM_], ms[DIM_], iss[DIM_], gs[DIM_], bs[DIM_];
    for (int c = threadIdx.x; c < DIM_; c += blockDim.x) {
        w3s[c] = W3[c];
        float m = sum2[c] * invE;
        float v = sumsq2[c] * invE - m * m;
        ms[c] = m;
        iss[c] = rsqrtf(v + EPS_);
        gs[c] = gamma2[c];
        bs[c] = beta2[c];
    }
    __syncthreads();
    int d = blockIdx.x * blockDim.x + threadIdx.x;
    if (d >= M) return;
    const float* y = Y2 + (size_t)d * DIM_;
    float acc = b3[0];
    for (int c = 0; c < DIM_; ++c) {
        float x = (y[c] - ms[c]) * iss[c] * gs[c] + bs[c];
        x = x > 0.0f ? x : 0.0f;
        acc += x * w3s[c];
    }
    sdis[d] = 1.0f / (1.0f + __expf(-acc));
}

__global__ void gather_scores_kernel(const int* __restrict__ dst, const float* __restrict__ sdis,
                                     float* __restrict__ out, int E) {
    int e = blockIdx.x * blockDim.x + threadIdx.x;
    if (e < E) out[e] = sdis[dst[e]];
}

// ---------------- launcher ----------------
extern "C" void kernel_launch(void* const* d_in, const int* in_sizes, int n_in,
                              void* d_out, int out_size, void* d_ws, size_t ws_size,
                              hipStream_t stream) {
    const float* h_drug    = (const float*)d_in[0];
    const float* W_gcn     = (const float*)d_in[2];
    const float* b_gcn     = (const float*)d_in[3];
    const float* W1        = (const float*)d_in[4];
    const float* b1        = (const float*)d_in[5];
    const float* gamma1    = (const float*)d_in[6];
    const float* beta1     = (const float*)d_in[7];
    const float* W2        = (const float*)d_in[8];
    const float* b2        = (const float*)d_in[9];
    const float* gamma2    = (const float*)d_in[10];
    const float* beta2     = (const float*)d_in[11];
    const float* W3        = (const float*)d_in[12];
    const float* b3        = (const float*)d_in[13];
    const int*   src       = (const int*)d_in[14];
    const int*   dst       = (const int*)d_in[15];
    float*       out       = (float*)d_out;

    float* ws       = (float*)d_ws;
    float* out_deg  = ws + OFF_OUTDEG;
    float* in_deg   = ws + OFF_INDEG;
    float* agg      = ws + OFF_AGG;
    float* sum1     = ws + OFF_SUM1;
    float* ssq1     = ws + OFF_SSQ1;
    float* sum2     = ws + OFF_SUM2;
    float* ssq2     = ws + OFF_SSQ2;
    float* norm_out = ws + OFF_NORMOUT;
    float* norm_in  = ws + OFF_NORMIN;
    float* dis_h    = ws + OFF_DISH;
    float* y1       = ws + OFF_Y1;
    float* y2       = ws + OFF_Y2;
    float* c1       = ws + OFF_C1;
    float* sdis     = ws + OFF_SDIS;

    const float invE = 1.0f / (float)E_;

    // 0) zero accumulators (every call: deterministic fresh state)
    zero_ws_kernel<<<(ZERO_N + 255) / 256, 256, 0, stream>>>(ws, ZERO_N);

    // 1) degrees
    degrees_kernel<<<(E_ + 255) / 256, 256, 0, stream>>>(src, dst, out_deg, in_deg, E_);

    // 2) norms = rsqrt(clip(deg, 1))
    make_norm_kernel<<<(ND_ + 255) / 256, 256, 0, stream>>>(out_deg, norm_out, ND_);
    make_norm_kernel<<<(NDIS_ + 255) / 256, 256, 0, stream>>>(in_deg, norm_in, NDIS_);

    // 3) scatter: agg[dst] += h_drug[src] * norm_out[src]   (drug half only; disease half is 0)
    {
        long total = (long)E_ * DIM_;
        scatter_agg_kernel<<<(unsigned)((total + 255) / 256), 256, 0, stream>>>(
            h_drug, src, dst, norm_out, agg, E_);
    }

    // 4) GEMM1: dis_h[10000,256] = (norm_in . agg)[10000,128] @ W_gcn[0:128,:] + b_gcn
    {
        int M = NDIS_, N = 256, K = 128;
        int tiles = (M / 16) * (N / 16);
        gemm16_wmma_kernel<<<(tiles + 7) / 8, 256, 0, stream>>>(agg, W_gcn, b_gcn, norm_in, dis_h, M, N, K);
    }

    // 5) c1 = b_gcn @ W1[0:256,:] + b1   (constant drug-half of every edge row)
    compute_c1_kernel<<<1, 256, 0, stream>>>(b_gcn, W1, b1, c1);

    // 6) GEMM2: y1[10000,256] = dis_h @ W1[256:512,:] + c1
    {
        int M = NDIS_, N = 256, K = 256;
        int tiles = (M / 16) * (N / 16);
        gemm16_wmma_kernel<<<(tiles + 7) / 8, 256, 0, stream>>>(dis_h, W1 + 256 * 256, c1, nullptr, y1, M, N, K);
    }

    // 7) BN1 stats (edge-batch stats == in_deg-weighted disease stats), then in-place BN+ReLU
    bn_stats_kernel<<<(NDIS_ + 99) / 100, 256, 0, stream>>>(y1, in_deg, sum1, ssq1, NDIS_, 256, 100);
    {
        long total = (long)NDIS_ * 256;
        bn_apply_relu_kernel<<<(unsigned)((total + 255) / 256), 256, 0, stream>>>(
            y1, sum1, ssq1, gamma1, beta1, invE, total, 256);
    }

    // 8) GEMM3: y2[10000,128] = x1 @ W2 + b2
    {
        int M = NDIS_, N = 128, K = 256;
        int tiles = (M / 16) * (N / 16);
        gemm16_wmma_kernel<<<(tiles + 7) / 8, 256, 0, stream>>>(y1, W2, b2, nullptr, y2, M, N, K);
    }

    // 9) BN2 stats, then fused BN2+ReLU+W3-dot+sigmoid per disease
    bn_stats_kernel<<<(NDIS_ + 99) / 100, 128, 0, stream>>>(y2, in_deg, sum2, ssq2, NDIS_, 128, 100);
    final_score_kernel<<<(NDIS_ + 255) / 256, 256, 0, stream>>>(
        y2, sum2, ssq2, gamma2, beta2, W3, b3, invE, sdis, NDIS_);

    // 10) score[e] = score_dis[dst[e]]
    gather_scores_kernel<<<(E_ + 255) / 256, 256, 0, stream>>>(dst, sdis, out, E_);
}